// SpanRepLayer_65678639890662
// MI455X (gfx1250) — compile-verified
//
#include <hip/hip_runtime.h>
#include <stdint.h>

// Problem constants (from the reference): B=4, S=1024, H=768, N=4096, W=16, win=1.
namespace {
constexpr int kB  = 4;
constexpr int kS  = 1024;
constexpr int kH  = 768;
constexpr int kN  = 4096;
constexpr int kW  = 16;
constexpr int kHC = kH / 4;           // 192 float4 chunks per token row
constexpr float kNeg = -1e30f;
}

typedef float v4f __attribute__((ext_vector_type(4)));
typedef int   v4i __attribute__((ext_vector_type(4)));

// gfx1250 async global->LDS path (ASYNCcnt-tracked). The builtin's signature
// (from the round-1 diagnostic) takes an AS1 int4* source and (by symmetry /
// ISA operand table) an AS3 int4* LDS destination, plus imm offset + cpol.
#if defined(__gfx1250__) && __has_builtin(__builtin_amdgcn_global_load_async_to_lds_b128)
#define SPAN_ASYNC 1
typedef __attribute__((address_space(1))) v4i gv4i_t;
typedef __attribute__((address_space(3))) v4i lv4i_t;

// Generic->AS1: global addresses pass through the generic aperture unchanged.
static __device__ inline gv4i_t* to_global_v4i(const void* p) {
  return (gv4i_t*)(uintptr_t)p;
}
// Generic->AS3: low 32 bits of a generic pointer to __shared__ are the
// wave-relative LDS byte offset (ISA 10.2: the LDS address is the low 32 bits);
// AS3 pointers are 32-bit, so the truncating integer cast is exact.
static __device__ inline lv4i_t* to_lds_v4i(void* p) {
  return (lv4i_t*)(uint32_t)(uintptr_t)p;
}
#else
#define SPAN_ASYNC 0
#endif

static __device__ inline v4f vmax4(v4f a, v4f b) {
  v4f r;
  r.x = __builtin_fmaxf(a.x, b.x);
  r.y = __builtin_fmaxf(a.y, b.y);
  r.z = __builtin_fmaxf(a.z, b.z);
  r.w = __builtin_fmaxf(a.w, b.w);
  return r;
}

// One span per 192-thread block (6 wave32s). Thread c owns float4 chunk c of H.
// All control flow is uniform per block (span params come from blockIdx).
__global__ __launch_bounds__(kHC) void span_rep_kernel(
    const float* __restrict__ token_reps,   // (B, S, H)
    const int2*  __restrict__ span_ids,     // (B, N) of {start, end}
    float*       __restrict__ out) {        // (B, N, 3H)
  const int span = blockIdx.x;              // b * kN + n
  const int b    = span / kN;
  const int c    = threadIdx.x;             // chunk 0..191

  const int2 se  = span_ids[span];
  const int start = se.x;
  const int end   = se.y;
  const int len   = end - start;

  v4f* __restrict__ op = (v4f*)(out + (size_t)span * (3 * kH));

  if (len <= 0) {
    // Empty span -> zeros(3H). Non-temporal: output is write-once streaming.
    v4f z = {0.0f, 0.0f, 0.0f, 0.0f};
    __builtin_nontemporal_store(z, op + c);
    __builtin_nontemporal_store(z, op + kHC + c);
    __builtin_nontemporal_store(z, op + 2 * kHC + c);
    return;
  }

  const v4f* __restrict__ base = (const v4f*)(token_reps + (size_t)b * kS * kH);
  const int ninner = len - 2;               // valid inner rows (win=1), <= 14

#if SPAN_ASYNC
  // Kick off async DMA of the inner-window rows into LDS (<= 14 rows * 3KB).
  // Fire-and-forget per lane; overlapped with the direct start/end row work.
  __shared__ v4f lbuf[(kW - 2) * kHC];      // 14 * 192 * 16B = 42 KB
  for (int r = 0; r < ninner; ++r) {
    const v4f* src = base + (size_t)(start + 1 + r) * kHC + c;
    __builtin_amdgcn_global_load_async_to_lds_b128(
        to_global_v4i(src),
        to_lds_v4i(&lbuf[r * kHC + c]),
        /*offset=*/0, /*cpol=*/0);
  }
#endif

  // start / end reps: single-row gathers (L2-resident), b128 loads.
  const v4f s = base[(size_t)start * kHC + c];
  const v4f e = (len > 1) ? base[(size_t)(end - 1) * kHC + c] : s;

  __builtin_nontemporal_store(s, op + c);               // start_rep
  __builtin_nontemporal_store(e, op + 2 * kHC + c);     // end_rep

  v4f inner = s;                                        // len <= 2*win fallback
  if (ninner > 0) {
    v4f acc = {kNeg, kNeg, kNeg, kNeg};
#if SPAN_ASYNC
    // Drain our wave's async copies; each lane only reads bytes it copied,
    // so no cross-wave barrier is required.
#if __has_builtin(__builtin_amdgcn_s_wait_asynccnt)
    __builtin_amdgcn_s_wait_asynccnt(0);
#else
    asm volatile("s_wait_asynccnt 0" ::: "memory");
#endif
    for (int r = 0; r < ninner; ++r) {
      acc = vmax4(acc, lbuf[r * kHC + c]);              // ds_load_b128
    }
#else
    for (int r = 0; r < ninner; ++r) {
      acc = vmax4(acc, base[(size_t)(start + 1 + r) * kHC + c]);
    }
#endif
    inner = acc;
  }
  __builtin_nontemporal_store(inner, op + kHC + c);     // inner_rep
}

extern "C" void kernel_launch(void* const* d_in, const int* in_sizes, int n_in,
                              void* d_out, int out_size, void* d_ws, size_t ws_size,
                              hipStream_t stream) {
  (void)in_sizes; (void)n_in; (void)out_size; (void)d_ws; (void)ws_size;
  const float* token_reps = (const float*)d_in[0];   // (B, S, H) f32
  const int2*  span_ids   = (const int2*)d_in[1];    // (B, N, 2) i32
  float*       out        = (float*)d_out;           // (B, N, 3H) f32

  dim3 grid(kB * kN);     // 16384 spans, one per block
  dim3 block(kHC);        // 192 threads = 6 wave32s
  span_rep_kernel<<<grid, block, 0, stream>>>(token_reps, span_ids, out);
}